// DGCNN_partseg_68436008894557
// MI455X (gfx1250) — compile-verified
//
#include <hip/hip_runtime.h>
#include <hip/hip_bf16.h>

// ---------------------------------------------------------------------------
// DGCNN part-segmentation forward for gfx1250 (MI455X), fp32 throughout.
// All 1x1-conv / FC GEMMs run on V_WMMA_F32_16X16X4_F32 with a 64x64 block
// tile (4 waves x [16 rows x 64 cols], 4 accumulators per wave, shared A
// fragment). Edge-feature construction (knn gather, nbr-ctr / ctr concat) is
// fused into the GEMM B-tile load so the 336MB (B,2C,N,K) tensor is never
// materialized. LDS tiles padded to 18-float rows: conflict-free on 64 banks
// while preserving 8-byte alignment for paired ds_load_b64 fragment reads.
// ---------------------------------------------------------------------------

typedef float v2f __attribute__((ext_vector_type(2)));
typedef float v8f __attribute__((ext_vector_type(8)));

#define BB 8
#define NN 2048
#define KK 40
#define NKL (NN * KK)           // 81920
#define LRELU_SLOPE 0.2f
#define BN_EPS 1e-5f

// ---------------------------------------------------------------------------
// k-NN: one block per (b,n). Distance row in LDS, 40 rounds of block argmax.
// top_k is over pdist = -||x_n - x_m||^2 (largest first), ties -> lower index.
// ---------------------------------------------------------------------------
__global__ void __launch_bounds__(256)
knn_kernel(const float* __restrict__ x, int* __restrict__ idx,
           int C, int Kn, int cstride, int coff)
{
    __shared__ float dist[NN];
    __shared__ float xn[128];
    __shared__ float rv[256];
    __shared__ int   ri[256];

    int bn = blockIdx.x;
    int b = bn / NN, n = bn % NN;
    const float* xb = x + (size_t)b * cstride * NN + (size_t)coff * NN;

    for (int c = threadIdx.x; c < C; c += 256) xn[c] = xb[(size_t)c * NN + n];
    __syncthreads();

    for (int m = threadIdx.x; m < NN; m += 256) {
        float s = 0.f;
        for (int c = 0; c < C; ++c) {
            float d = xb[(size_t)c * NN + m] - xn[c];
            s += d * d;
        }
        dist[m] = -s;
    }
    __syncthreads();

    for (int r = 0; r < Kn; ++r) {
        float bv = -3.4e38f; int bi = 0;
        for (int m = threadIdx.x; m < NN; m += 256) {
            float v = dist[m];
            if (v > bv || (v == bv && m < bi)) { bv = v; bi = m; }
        }
        rv[threadIdx.x] = bv; ri[threadIdx.x] = bi;
        __syncthreads();
        for (int s = 128; s > 0; s >>= 1) {
            if (threadIdx.x < s) {
                float ov = rv[threadIdx.x + s]; int oi = ri[threadIdx.x + s];
                if (ov > rv[threadIdx.x] ||
                    (ov == rv[threadIdx.x] && oi < ri[threadIdx.x])) {
                    rv[threadIdx.x] = ov; ri[threadIdx.x] = oi;
                }
            }
            __syncthreads();
        }
        if (threadIdx.x == 0) {
            idx[(size_t)bn * Kn + r] = ri[0];
            dist[ri[0]] = -3.4e38f;
        }
        __syncthreads();
    }
}

// ---------------------------------------------------------------------------
// WMMA GEMM:  out(o,j) = sum_c W[o,c] * X(c,j)
//   mode 0 (dense): X(c,j) = act[b*cstride*L + (coff+c)*L + pos], j=(b,pos)
//   mode 1 (edge):  j=(b, n*Kn+k); c<Cc: x[.,c,idx]-x[.,c,n]; else x[.,c-Cc,n]
// Block: 128 threads = 4 waves. Block tile 64 rows x 64 cols; each wave owns
// 16 rows x 64 cols (4 v8f accumulators, A fragment reused across the 4
// column fragments). K-chunked by 16 via padded LDS tiles.
// ---------------------------------------------------------------------------
#define GTM 64
#define GTN 64
#define GTK 16
#define XPAD 18   // row pitch (floats): conflict-free on 64 banks, 8B aligned

__global__ void __launch_bounds__(128)
gemm_wmma(const float* __restrict__ W, const float* __restrict__ X,
          const int* __restrict__ idx, float* __restrict__ out,
          int O, int Cin, int Ncols, int L,
          int mode, int Cc, int Kn, int cstride, int coff)
{
    __shared__ float Ws[GTM * XPAD];   // Ws[m*XPAD + k]
    __shared__ float Xs[GTN * XPAD];   // Xs[n*XPAD + k]

    int tid  = threadIdx.x;
    int lane = tid & 31;
    int wave = tid >> 5;
    int col0 = blockIdx.x * GTN;
    int row0 = blockIdx.y * GTM;

    v8f acc0 = {0.f,0.f,0.f,0.f,0.f,0.f,0.f,0.f};
    v8f acc1 = acc0, acc2 = acc0, acc3 = acc0;

    for (int k0 = 0; k0 < Cin; k0 += GTK) {
        // stage W chunk (64 rows x 16 k); k fastest across threads -> coalesced
        for (int i = tid; i < GTM * GTK; i += 128) {
            int m = i >> 4, k = i & 15;
            int o = row0 + m, c = k0 + k;
            Ws[m * XPAD + k] = (o < O && c < Cin) ? W[(size_t)o * Cin + c] : 0.f;
        }
        // prefetch next W chunk (global_prefetch_b8)
        if (k0 + GTK < Cin) {
            int pm = row0 + (tid >> 1);
            if (pm < O)
                __builtin_prefetch(&W[(size_t)pm * Cin + k0 + GTK + (tid & 1) * 8], 0, 0);
        }
        // stage X chunk (64 cols x 16 k); col index fastest across threads so
        // global loads (and edge-mode idx gathers) are coalesced.
        for (int i = tid; i < GTN * GTK; i += 128) {
            int n = i & 63, k = i >> 6;
            int j = col0 + n, c = k0 + k;
            float v = 0.f;
            if (j < Ncols && c < Cin) {
                int b = j / L, pos = j - b * L;
                if (mode == 0) {
                    v = X[(size_t)b * cstride * L + (size_t)(coff + c) * L + pos];
                } else {
                    int pn = pos / Kn, pk = pos - pn * Kn;
                    const float* xb = X + (size_t)b * cstride * NN;
                    if (c < Cc) {
                        int nb = idx[((size_t)b * NN + pn) * Kn + pk];
                        v = xb[(size_t)(coff + c) * NN + nb]
                          - xb[(size_t)(coff + c) * NN + pn];
                    } else {
                        v = xb[(size_t)(coff + c - Cc) * NN + pn];
                    }
                }
            }
            Xs[n * XPAD + k] = v;
        }
        __syncthreads();

        // 16 x V_WMMA_F32_16X16X4_F32 per chunk per wave (A reused x4)
        {
            int mrow = lane & 15;
            int kb   = (lane >> 4) * 2;      // lanes 0-15: K=0,1 ; 16-31: K=2,3
            const float* wsB = &Ws[(wave * 16 + mrow) * XPAD + kb];
            const float* xsB = &Xs[(lane & 15) * XPAD + kb];
#pragma unroll
            for (int kk = 0; kk < GTK; kk += 4) {
                v2f a, b0, b1, b2, b3;
                a.x  = wsB[kk];                a.y  = wsB[kk + 1];
                b0.x = xsB[kk];                b0.y = xsB[kk + 1];
                b1.x = xsB[16 * XPAD + kk];    b1.y = xsB[16 * XPAD + kk + 1];
                b2.x = xsB[32 * XPAD + kk];    b2.y = xsB[32 * XPAD + kk + 1];
                b3.x = xsB[48 * XPAD + kk];    b3.y = xsB[48 * XPAD + kk + 1];
                acc0 = __builtin_amdgcn_wmma_f32_16x16x4_f32(
                           false, a, false, b0, (short)0, acc0, false, false);
                acc1 = __builtin_amdgcn_wmma_f32_16x16x4_f32(
                           false, a, false, b1, (short)0, acc1, false, false);
                acc2 = __builtin_amdgcn_wmma_f32_16x16x4_f32(
                           false, a, false, b2, (short)0, acc2, false, false);
                acc3 = __builtin_amdgcn_wmma_f32_16x16x4_f32(
                           false, a, false, b3, (short)0, acc3, false, false);
            }
        }
        __syncthreads();
    }

    // store: VGPR r -> row r (lanes 0-15) / row r+8 (lanes 16-31)
    int mbase = row0 + wave * 16 + (lane >> 4) * 8;
    int nloc  = lane & 15;
    v8f accs[4] = {acc0, acc1, acc2, acc3};
#pragma unroll
    for (int f = 0; f < 4; ++f) {
        int j = col0 + f * 16 + nloc;
        if (j < Ncols) {
            int b = j / L, pos = j - b * L;
#pragma unroll
            for (int r = 0; r < 8; ++r) {
                int o = mbase + r;
                if (o < O)
                    out[(size_t)b * O * L + (size_t)o * L + pos] = accs[f][r];
            }
        }
    }
}

// ---------------------------------------------------------------------------
// per-channel sum / sum-of-squares (BN stats), block reduce + atomics
// ---------------------------------------------------------------------------
__global__ void __launch_bounds__(256)
channel_stats(const float* __restrict__ x, float* __restrict__ sums,
              int B, int C, int L)
{
    __shared__ float ss[256], sq[256];
    int c = blockIdx.x;
    int stride = blockDim.x * gridDim.y;
    int start  = blockIdx.y * blockDim.x + threadIdx.x;
    float s = 0.f, q = 0.f;
    for (int b = 0; b < B; ++b) {
        const float* p = x + ((size_t)b * C + c) * (size_t)L;
        for (int i = start; i < L; i += stride) {
            float v = p[i]; s += v; q += v * v;
        }
    }
    ss[threadIdx.x] = s; sq[threadIdx.x] = q;
    __syncthreads();
    for (int d = 128; d > 0; d >>= 1) {
        if (threadIdx.x < d) {
            ss[threadIdx.x] += ss[threadIdx.x + d];
            sq[threadIdx.x] += sq[threadIdx.x + d];
        }
        __syncthreads();
    }
    if (threadIdx.x == 0) {
        atomicAdd(&sums[c], ss[0]);
        atomicAdd(&sums[C + c], sq[0]);
    }
}

__global__ void bn_lrelu(float* __restrict__ x, const float* __restrict__ sums,
                         const float* __restrict__ g, const float* __restrict__ bta,
                         int C, int L, size_t total, float invcnt, int relu)
{
    size_t t = (size_t)blockIdx.x * blockDim.x + threadIdx.x;
    if (t >= total) return;
    int c = (int)((t / (size_t)L) % (size_t)C);
    float m  = sums[c] * invcnt;
    float vv = sums[C + c] * invcnt - m * m;
    float r  = rsqrtf(vv + BN_EPS);
    float y  = (x[t] - m) * r * g[c] + bta[c];
    if (relu) y = (y >= 0.f) ? y : LRELU_SLOPE * y;
    x[t] = y;
}

__global__ void add_bias(float* __restrict__ x, const float* __restrict__ bias,
                         int C, int L, size_t total)
{
    size_t t = (size_t)blockIdx.x * blockDim.x + threadIdx.x;
    if (t >= total) return;
    int c = (int)((t / (size_t)L) % (size_t)C);
    x[t] += bias[c];
}

// (B,C,N,K) -> max over K -> write into (B,Ctot,N) at channel offset coff
__global__ void max_over_k(const float* __restrict__ in, float* __restrict__ out,
                           int C, int Kn, int Ctot, int coff)
{
    int t = blockIdx.x * blockDim.x + threadIdx.x;
    int total = BB * C * NN;
    if (t >= total) return;
    int n = t % NN; int rest = t / NN;
    int c = rest % C; int b = rest / C;
    const float* p = in + (((size_t)b * C + c) * NN + n) * Kn;
    float m = p[0];
    for (int k = 1; k < Kn; ++k) m = fmaxf(m, p[k]);
    out[((size_t)b * Ctot + (coff + c)) * NN + n] = m;
}

// (B,C,L) -> max over L -> (B,C); one block per (b,c)
__global__ void __launch_bounds__(256)
max_over_l(const float* __restrict__ in, float* __restrict__ out, int L)
{
    __shared__ float red[256];
    int bc = blockIdx.x;
    const float* p = in + (size_t)bc * L;
    float m = -3.4e38f;
    for (int i = threadIdx.x; i < L; i += 256) m = fmaxf(m, p[i]);
    red[threadIdx.x] = m;
    __syncthreads();
    for (int d = 128; d > 0; d >>= 1) {
        if (threadIdx.x < d)
            red[threadIdx.x] = fmaxf(red[threadIdx.x], red[threadIdx.x + d]);
        __syncthreads();
    }
    if (threadIdx.x == 0) out[bc] = red[0];
}

// x' [b,d,n] = sum_c x[b,c,n] * t9[b*9 + c*3 + d]
__global__ void transform_apply(const float* __restrict__ x,
                                const float* __restrict__ t9,
                                float* __restrict__ out)
{
    int t = blockIdx.x * blockDim.x + threadIdx.x;
    if (t >= BB * 3 * NN) return;
    int n = t % NN; int d = (t / NN) % 3; int b = t / (3 * NN);
    float s = 0.f;
    for (int c = 0; c < 3; ++c)
        s += x[((size_t)b * 3 + c) * NN + n] * t9[b * 9 + c * 3 + d];
    out[((size_t)b * 3 + d) * NN + n] = s;
}

// h(B,1280,N): [0,1024)=gmax bcast, [1024,1088)=lab bcast, [1088,1280)=x0
__global__ void build_h(const float* __restrict__ gmax, const float* __restrict__ lab,
                        const float* __restrict__ x0, float* __restrict__ h)
{
    size_t t = (size_t)blockIdx.x * blockDim.x + threadIdx.x;
    size_t total = (size_t)BB * 1280 * NN;
    if (t >= total) return;
    int n = (int)(t % NN); int c = (int)((t / NN) % 1280); int b = (int)(t / (1280ull * NN));
    float v;
    if (c < 1024)      v = gmax[b * 1024 + c];
    else if (c < 1088) v = lab[b * 64 + (c - 1024)];
    else               v = x0[((size_t)b * 192 + (c - 1088)) * NN + n];
    h[t] = v;
}

// ---------------------------------------------------------------------------
// host orchestration
// ---------------------------------------------------------------------------
static void launch_gemm(const float* W, const float* X, const int* idx, float* out,
                        int O, int Cin, int Ncols, int L, int mode, int Cc,
                        int cstride, int coff, hipStream_t s)
{
    dim3 g((Ncols + GTN - 1) / GTN, (O + GTM - 1) / GTM);
    gemm_wmma<<<g, 128, 0, s>>>(W, X, idx, out, O, Cin, Ncols, L, mode, Cc, KK, cstride, coff);
}

static void bn_block(float* x, float* stats, const float* g, const float* b,
                     int B, int C, int L, int relu, hipStream_t s)
{
    hipMemsetAsync(stats, 0, 2 * (size_t)C * sizeof(float), s);
    dim3 gs((unsigned)C, (L > 4096) ? 64u : 4u);
    channel_stats<<<gs, 256, 0, s>>>(x, stats, B, C, L);
    size_t tot = (size_t)B * C * L;
    bn_lrelu<<<(unsigned)((tot + 255) / 256), 256, 0, s>>>(
        x, stats, g, b, C, L, tot, 1.f / ((float)B * (float)L), relu);
}

extern "C" void kernel_launch(void* const* d_in, const int* in_sizes, int n_in,
                              void* d_out, int out_size, void* d_ws, size_t ws_size,
                              hipStream_t stream)
{
    (void)in_sizes; (void)n_in; (void)out_size; (void)ws_size;
    const float* x_in   = (const float*)d_in[0];   // (8,3,2048)
    const float* labels = (const float*)d_in[1];   // (8,16)
    const float* P[64];
    for (int i = 2; i < 51; ++i) P[i] = (const float*)d_in[i];

    // workspace layout (float offsets)
    float* ws = (float*)d_ws;
    int*   idx = (int*)d_ws;                       // B*N*K ints
    const size_t off_xT   = 655360;                // (8,3,2048)
    const size_t off_bufA = off_xT   + 49152;      // (8, 64, N*K) = 41.9M
    const size_t off_bufB = off_bufA + (size_t)BB * 64 * NKL;   // (8,128,N*K) = 83.9M
    const size_t off_tnp  = off_bufB + (size_t)BB * 128 * NKL;  // (8,128,N)
    const size_t off_gbuf = off_tnp  + (size_t)BB * 128 * NN;   // (8,1024,N)
    const size_t off_x0   = off_gbuf + (size_t)BB * 1024 * NN;  // (8,192,N)
    const size_t off_h    = off_x0   + (size_t)BB * 192 * NN;   // (8,1280,N)
    const size_t off_p1   = off_h    + (size_t)BB * 1280 * NN;  // (8,256,N)
    const size_t off_p2   = off_p1   + (size_t)BB * 256 * NN;   // (8,256,N)
    const size_t off_hvec = off_p2   + (size_t)BB * 256 * NN;   // (8,1024)
    const size_t off_fc1  = off_hvec + BB * 1024;               // (8,512)
    const size_t off_fc2  = off_fc1  + BB * 512;                // (8,256)
    const size_t off_t9   = off_fc2  + BB * 256;                // (8,9) padded
    const size_t off_gmax = off_t9   + 128;                     // (8,1024)
    const size_t off_lab  = off_gmax + BB * 1024;               // (8,64)
    const size_t off_st   = off_lab  + BB * 64;                 // stats 2*2048
    float* xT   = ws + off_xT;
    float* bufA = ws + off_bufA;
    float* bufB = ws + off_bufB;
    float* tnp  = ws + off_tnp;
    float* gbuf = ws + off_gbuf;
    float* x0   = ws + off_x0;
    float* hbuf = ws + off_h;
    float* p1   = ws + off_p1;
    float* p2   = ws + off_p2;
    float* hvec = ws + off_hvec;
    float* fc1  = ws + off_fc1;
    float* fc2  = ws + off_fc2;
    float* t9   = ws + off_t9;
    float* gmax = ws + off_gmax;
    float* lab  = ws + off_lab;
    float* st   = ws + off_st;

    const int NCE = BB * NKL;   // edge-conv column count = 655360
    const int NCP = BB * NN;    // point-conv column count = 16384

    // ---- transform net ----
    knn_kernel<<<BB * NN, 256, 0, stream>>>(x_in, idx, 3, KK, 3, 0);
    launch_gemm(P[2], x_in, idx, bufA, 64, 6, NCE, NKL, 1, 3, 3, 0, stream);   // tn.c1
    bn_block(bufA, st, P[3], P[4], BB, 64, NKL, 1, stream);
    launch_gemm(P[5], bufA, 0, bufB, 128, 64, NCE, NKL, 0, 0, 64, 0, stream);  // tn.c2
    bn_block(bufB, st, P[6], P[7], BB, 128, NKL, 1, stream);
    max_over_k<<<(BB * 128 * NN + 255) / 256, 256, 0, stream>>>(bufB, tnp, 128, KK, 128, 0);
    launch_gemm(P[8], tnp, 0, gbuf, 1024, 128, NCP, NN, 0, 0, 128, 0, stream); // tn.c3
    bn_block(gbuf, st, P[9], P[10], BB, 1024, NN, 1, stream);
    max_over_l<<<BB * 1024, 256, 0, stream>>>(gbuf, hvec, NN);
    launch_gemm(P[11], hvec, 0, fc1, 512, 1024, BB, 1, 0, 0, 1024, 0, stream); // l1
    bn_block(fc1, st, P[12], P[13], BB, 512, 1, 1, stream);
    launch_gemm(P[14], fc1, 0, fc2, 256, 512, BB, 1, 0, 0, 512, 0, stream);    // l2
    bn_block(fc2, st, P[15], P[16], BB, 256, 1, 1, stream);
    launch_gemm(P[17], fc2, 0, t9, 9, 256, BB, 1, 0, 0, 256, 0, stream);       // tw
    add_bias<<<1, 128, 0, stream>>>(t9, P[18], 9, 1, (size_t)BB * 9);
    transform_apply<<<(BB * 3 * NN + 255) / 256, 256, 0, stream>>>(x_in, t9, xT);

    // ---- edge conv 1 ----
    knn_kernel<<<BB * NN, 256, 0, stream>>>(xT, idx, 3, KK, 3, 0);
    launch_gemm(P[19], xT, idx, bufA, 64, 6, NCE, NKL, 1, 3, 3, 0, stream);
    bn_block(bufA, st, P[20], P[21], BB, 64, NKL, 1, stream);
    launch_gemm(P[22], bufA, 0, bufB, 64, 64, NCE, NKL, 0, 0, 64, 0, stream);
    bn_block(bufB, st, P[23], P[24], BB, 64, NKL, 1, stream);
    max_over_k<<<(BB * 64 * NN + 255) / 256, 256, 0, stream>>>(bufB, x0, 64, KK, 192, 0);

    // ---- edge conv 2 ----
    knn_kernel<<<BB * NN, 256, 0, stream>>>(x0, idx, 64, KK, 192, 0);
    launch_gemm(P[25], x0, idx, bufA, 64, 128, NCE, NKL, 1, 64, 192, 0, stream);
    bn_block(bufA, st, P[26], P[27], BB, 64, NKL, 1, stream);
    launch_gemm(P[28], bufA, 0, bufB, 64, 64, NCE, NKL, 0, 0, 64, 0, stream);
    bn_block(bufB, st, P[29], P[30], BB, 64, NKL, 1, stream);
    max_over_k<<<(BB * 64 * NN + 255) / 256, 256, 0, stream>>>(bufB, x0, 64, KK, 192, 64);

    // ---- edge conv 3 ----
    knn_kernel<<<BB * NN, 256, 0, stream>>>(x0, idx, 64, KK, 192, 64);
    launch_gemm(P[31], x0, idx, bufA, 64, 128, NCE, NKL, 1, 64, 192, 64, stream);
    bn_block(bufA, st, P[32], P[33], BB, 64, NKL, 1, stream);
    max_over_k<<<(BB * 64 * NN + 255) / 256, 256, 0, stream>>>(bufA, x0, 64, KK, 192, 128);

    // ---- global feature ----
    launch_gemm(P[34], x0, 0, gbuf, 1024, 192, NCP, NN, 0, 0, 192, 0, stream); // cx
    bn_block(gbuf, st, P[35], P[36], BB, 1024, NN, 1, stream);
    max_over_l<<<BB * 1024, 256, 0, stream>>>(gbuf, gmax, NN);
    launch_gemm(P[37], labels, 0, lab, 64, 16, BB, 1, 0, 0, 16, 0, stream);    // cl
    bn_block(lab, st, P[38], P[39], BB, 64, 1, 1, stream);
    build_h<<<(unsigned)(((size_t)BB * 1280 * NN + 255) / 256), 256, 0, stream>>>(gmax, lab, x0, hbuf);

    // ---- decoder ----
    launch_gemm(P[40], hbuf, 0, p1, 256, 1280, NCP, NN, 0, 0, 1280, 0, stream);
    bn_block(p1, st, P[41], P[42], BB, 256, NN, 1, stream);
    launch_gemm(P[43], p1, 0, p2, 256, 256, NCP, NN, 0, 0, 256, 0, stream);
    bn_block(p2, st, P[44], P[45], BB, 256, NN, 1, stream);
    launch_gemm(P[46], p2, 0, p1, 128, 256, NCP, NN, 0, 0, 256, 0, stream);
    bn_block(p1, st, P[47], P[48], BB, 128, NN, 1, stream);
    launch_gemm(P[49], p1, 0, (float*)d_out, 50, 128, NCP, NN, 0, 0, 128, 0, stream);
    add_bias<<<(unsigned)(((size_t)BB * 50 * NN + 255) / 256), 256, 0, stream>>>(
        (float*)d_out, P[50], 50, NN, (size_t)BB * 50 * NN);
}